// Dual_Non_Local_Block_76364518523099
// MI455X (gfx1250) — compile-verified
//
#include <hip/hip_runtime.h>
#include <hip/hip_bf16.h>

#define B_  4
#define C_  192
#define H_  256
#define W_  256
#define MID_ 96
#define HW_ (H_ * W_)

typedef float v2f __attribute__((ext_vector_type(2)));
typedef float v8f __attribute__((ext_vector_type(8)));

__device__ __forceinline__ float waveSum(float v) {
#pragma unroll
    for (int off = 16; off; off >>= 1) v += __shfl_xor(v, off, 32);
    return v;
}
__device__ __forceinline__ float waveMax(float v) {
#pragma unroll
    for (int off = 16; off; off >>= 1) v = fmaxf(v, __shfl_xor(v, off, 32));
    return v;
}

// ---------------------------------------------------------------------------
// Pass 1: x1[b,c,h] = mean_w x, x2[b,c,w] = mean_h x.  One b128 read of x.
// One block per (b,c) plane (768 blocks x 256 threads).
// Thread layout: g = t>>6 selects one of 4 interleaved rows per iteration,
// q = t&63 selects a float4 column group (cols 4q..4q+3).
// Deterministic: fixed-order LDS partial combines (no float atomics).
// ---------------------------------------------------------------------------
__global__ void k_means(const float* __restrict__ x,
                        float* __restrict__ x1, float* __restrict__ x2) {
    __shared__ float rowpart[H_ * 2];    // 2 wave-partials per row
    __shared__ float colpart[4 * W_];    // 4 group-partials per column
    const int t = threadIdx.x;
    const int g = t >> 6;                // row group 0..3
    const int q = t & 63;                // float4 column index
    const int wslot = (t >> 5) & 1;      // which of the 2 waves covering a row
    const float4* p4 = (const float4*)(x + (size_t)blockIdx.x * HW_);
    float c0 = 0.f, c1 = 0.f, c2 = 0.f, c3 = 0.f;
    for (int h = 0; h < H_ / 4; ++h) {
        const int row = h * 4 + g;
        const float4 v = p4[row * (W_ / 4) + q];
        c0 += v.x; c1 += v.y; c2 += v.z; c3 += v.w;
        float r = waveSum(v.x + v.y + v.z + v.w);
        if ((t & 31) == 0) rowpart[row * 2 + wslot] = r;
    }
    colpart[g * W_ + q * 4 + 0] = c0;
    colpart[g * W_ + q * 4 + 1] = c1;
    colpart[g * W_ + q * 4 + 2] = c2;
    colpart[g * W_ + q * 4 + 3] = c3;
    __syncthreads();
    // thread t = row index for x1, column index for x2 (fixed combine order)
    x1[(size_t)blockIdx.x * H_ + t] =
        (rowpart[t * 2] + rowpart[t * 2 + 1]) * (1.f / (float)W_);
    x2[(size_t)blockIdx.x * W_ + t] =
        (colpart[t] + colpart[W_ + t] + colpart[2 * W_ + t] + colpart[3 * W_ + t]) *
        (1.f / (float)H_);
}

// ---------------------------------------------------------------------------
// Pass 2: 1-D conv (K=5, pad 2) over pooled axes: out[b,m,t] (tiny).
// One block per (b,m): 384 blocks x 256 threads.
// ---------------------------------------------------------------------------
__global__ void k_conv(const float* __restrict__ in,   // [B, C, 256]
                       const float* __restrict__ w,    // [MID, C, 5]
                       const float* __restrict__ bias, // [MID]
                       float* __restrict__ out) {      // [B, MID, 256]
    const int bm = blockIdx.x;
    const int b = bm / MID_, m = bm % MID_;
    const int t = threadIdx.x;
    const float* ip = in + (size_t)b * C_ * 256;
    const float* wp = w + (size_t)m * C_ * 5;
    float acc = bias[m];
    for (int c = 0; c < C_; ++c) {
        const float* row = ip + c * 256;
        const float* wk  = wp + c * 5;
#pragma unroll
        for (int k = 0; k < 5; ++k) {
            int pos = t + k - 2;
            float v = (pos >= 0 && pos < 256) ? row[pos] : 0.f;
            acc = fmaf(wk[k], v, acc);
        }
    }
    out[(size_t)bm * 256 + t] = acc;
}

// ---------------------------------------------------------------------------
// Pass 3: f1[b,h,w] = sum_m theta[b,m,h] * phi[b,m,w]  via V_WMMA_F32_16X16X4_F32.
// One 16x16 output tile per wave32; 8 waves per block; grid = B*256/8 = 128.
// A(16x4): lanes 0-15 hold K={0,1} in v0/v1, lanes 16-31 hold K={2,3}.
// B(4x16): v0 = rows K=0 (lanes 0-15) / K=2 (lanes 16-31); v1 = K=1 / K=3.
// D(16x16): VGPR i -> row i (lanes 0-15), row i+8 (lanes 16-31).
// ---------------------------------------------------------------------------
__global__ void k_f1_wmma(const float* __restrict__ theta,  // [B, MID, 256]
                          const float* __restrict__ phi,    // [B, MID, 256]
                          float* __restrict__ f1) {         // [B, 256, 256]
    const int gwave = (blockIdx.x * blockDim.x + threadIdx.x) >> 5;
    const int lane  = threadIdx.x & 31;
    const int b     = gwave >> 8;          // 256 tiles per batch
    const int tile  = gwave & 255;
    const int h0 = (tile >> 4) << 4;
    const int n0 = (tile & 15) << 4;
    const float* th = theta + (size_t)b * MID_ * 256;
    const float* ph = phi   + (size_t)b * MID_ * 256;
    const int lm = lane & 15;
    const int kq = (lane >> 4) << 1;       // 0 or 2
    v8f c = {0.f, 0.f, 0.f, 0.f, 0.f, 0.f, 0.f, 0.f};
    for (int k0 = 0; k0 < MID_; k0 += 4) {
        v2f a, bb;
        a.x  = th[(k0 + kq    ) * 256 + h0 + lm];
        a.y  = th[(k0 + kq + 1) * 256 + h0 + lm];
        bb.x = ph[(k0 + kq    ) * 256 + n0 + lm];
        bb.y = ph[(k0 + kq + 1) * 256 + n0 + lm];
        c = __builtin_amdgcn_wmma_f32_16x16x4_f32(false, a, false, bb,
                                                  (short)0, c, false, false);
    }
    float* o = f1 + (size_t)b * HW_;
    const int row0 = h0 + ((lane >> 4) << 3);
#pragma unroll
    for (int i = 0; i < 8; ++i)
        o[(row0 + i) * 256 + n0 + lm] = c[i];
}

// ---------------------------------------------------------------------------
// Pass 4: row softmax + spatial gate:
//   a[b,hw] = softmax_w(f1)[b,h,w] * (1 + sigmoid(w_c1*f1 + b_c1))
// One block per (b,h): 1024 blocks x 256 threads.
// ---------------------------------------------------------------------------
__global__ void k_attn(const float* __restrict__ f1,
                       const float* __restrict__ wc1p, const float* __restrict__ bc1p,
                       float* __restrict__ a) {
    __shared__ float l[8];
    const int bh = blockIdx.x;
    const int t  = threadIdx.x;
    const int lane = t & 31, wid = t >> 5;
    const float v = f1[(size_t)bh * 256 + t];
    // block max
    float m = waveMax(v);
    if (lane == 0) l[wid] = m;
    __syncthreads();
    if (t == 0) { float r = l[0]; for (int i = 1; i < 8; ++i) r = fmaxf(r, l[i]); l[0] = r; }
    __syncthreads();
    m = l[0];
    __syncthreads();
    // block sum of exp (fixed combine order)
    const float e = __expf(v - m);
    float s = waveSum(e);
    if (lane == 0) l[wid] = s;
    __syncthreads();
    if (t == 0) { float r = 0.f; for (int i = 0; i < 8; ++i) r += l[i]; l[0] = r; }
    __syncthreads();
    s = l[0];
    const float sig = 1.f / (1.f + __expf(-(wc1p[0] * v + bc1p[0])));
    a[(size_t)bh * 256 + t] = (e / s) * (1.f + sig);
}

// ---------------------------------------------------------------------------
// Pass 5: t[b,c] = sum_hw a[b,hw] * x[b,c,hw].  Second read of x (b128; a hits L2).
// One block per (b,c): 768 blocks x 256 threads.
// ---------------------------------------------------------------------------
__global__ void k_tred(const float* __restrict__ x, const float* __restrict__ a,
                       float* __restrict__ t) {
    __shared__ float l[8];
    const int bc = blockIdx.x;
    const int b  = bc / C_;
    const int tid = threadIdx.x;
    const float4* xp = (const float4*)(x + (size_t)bc * HW_);
    const float4* ap = (const float4*)(a + (size_t)b * HW_);
    float acc = 0.f;
    for (int i = 0; i < HW_ / 4 / 256; ++i) {        // 64 iterations
        const int idx = i * 256 + tid;
        if (i < 56) __builtin_prefetch(&xp[idx + 2048], 0, 3);  // global_prefetch_b8
        const float4 xv = xp[idx];
        const float4 av = ap[idx];
        acc = fmaf(xv.x, av.x, acc);
        acc = fmaf(xv.y, av.y, acc);
        acc = fmaf(xv.z, av.z, acc);
        acc = fmaf(xv.w, av.w, acc);
    }
    acc = waveSum(acc);
    if ((tid & 31) == 0) l[tid >> 5] = acc;
    __syncthreads();
    if (tid == 0) { float r = 0.f; for (int i = 0; i < 8; ++i) r += l[i]; t[bc] = r; }
}

// ---------------------------------------------------------------------------
// Pass 6: y = w_kxk @ t + 256*b_kxk ; scale[b,o] = 1 + sigmoid(w_c2 @ y + b_c2)
// Tiny: grid B, 192 threads.
// ---------------------------------------------------------------------------
__global__ void k_gate(const float* __restrict__ t,
                       const float* __restrict__ w_kxk, const float* __restrict__ b_kxk,
                       const float* __restrict__ w_c2,  const float* __restrict__ b_c2,
                       float* __restrict__ scale) {
    __shared__ float y[MID_];
    const int b = blockIdx.x;
    const int o = threadIdx.x;   // 0..191
    if (o < MID_) {
        float acc = 256.f * b_kxk[o];
        for (int c = 0; c < C_; ++c) acc = fmaf(w_kxk[o * C_ + c], t[b * C_ + c], acc);
        y[o] = acc;
    }
    __syncthreads();
    float acc = b_c2[o];
    for (int m = 0; m < MID_; ++m) acc = fmaf(w_c2[o * MID_ + m], y[m], acc);
    scale[b * C_ + o] = 1.f + 1.f / (1.f + __expf(-acc));
}

// ---------------------------------------------------------------------------
// Pass 7: out[b,c,h,w] = scale[b,c] * x[b,c,h,w]   (float4 streaming)
// ---------------------------------------------------------------------------
__global__ void k_out(const float* __restrict__ x, const float* __restrict__ scale,
                      float* __restrict__ out) {
    const size_t gid = (size_t)blockIdx.x * blockDim.x + threadIdx.x;  // vec4 index
    const int plane = (int)(gid >> 14);          // HW_/4 = 16384 vec4 per (b,c)
    const float s = scale[plane];
    const float4 v = ((const float4*)x)[gid];
    float4 r;
    r.x = v.x * s; r.y = v.y * s; r.z = v.z * s; r.w = v.w * s;
    ((float4*)out)[gid] = r;
}

extern "C" void kernel_launch(void* const* d_in, const int* in_sizes, int n_in,
                              void* d_out, int out_size, void* d_ws, size_t ws_size,
                              hipStream_t stream) {
    const float* x      = (const float*)d_in[0];
    const float* w_kx1  = (const float*)d_in[1];
    const float* b_kx1  = (const float*)d_in[2];
    const float* w_1xk  = (const float*)d_in[3];
    const float* b_1xk  = (const float*)d_in[4];
    const float* w_kxk  = (const float*)d_in[5];
    const float* b_kxk  = (const float*)d_in[6];
    const float* w_c1   = (const float*)d_in[7];
    const float* b_c1   = (const float*)d_in[8];
    const float* w_c2   = (const float*)d_in[9];
    const float* b_c2   = (const float*)d_in[10];
    float* out = (float*)d_out;

    float* ws = (float*)d_ws;
    float* x1    = ws;                               // B*C*H   = 196608
    float* x2    = x1 + (size_t)B_ * C_ * H_;        // 196608
    float* theta = x2 + (size_t)B_ * C_ * W_;        // B*MID*H = 98304
    float* phi   = theta + (size_t)B_ * MID_ * H_;   // 98304
    float* f1    = phi + (size_t)B_ * MID_ * W_;     // B*H*W   = 262144
    float* av    = f1 + (size_t)B_ * HW_;            // 262144
    float* tv    = av + (size_t)B_ * HW_;            // B*C     = 768
    float* scale = tv + (size_t)B_ * C_;             // 768

    k_means<<<B_ * C_, 256, 0, stream>>>(x, x1, x2);
    k_conv<<<B_ * MID_, 256, 0, stream>>>(x1, w_kx1, b_kx1, theta);
    k_conv<<<B_ * MID_, 256, 0, stream>>>(x2, w_1xk, b_1xk, phi);
    k_f1_wmma<<<(B_ * 256) / 8, 256, 0, stream>>>(theta, phi, f1);
    k_attn<<<B_ * H_, 256, 0, stream>>>(f1, w_c1, b_c1, av);
    k_tred<<<B_ * C_, 256, 0, stream>>>(x, av, tv);
    k_gate<<<B_, C_, 0, stream>>>(tv, w_kxk, b_kxk, w_c2, b_c2, scale);
    const size_t nvec4 = (size_t)B_ * C_ * HW_ / 4;  // 12,582,912
    k_out<<<(unsigned)(nvec4 / 256), 256, 0, stream>>>(x, scale, out);
}